// Attention_Correlation_weight_reshape_loss_15616501088817
// MI455X (gfx1250) — compile-verified
//
#include <hip/hip_runtime.h>

typedef __attribute__((ext_vector_type(2))) float v2f;
typedef __attribute__((ext_vector_type(4))) float v4f;
typedef __attribute__((ext_vector_type(8))) float v8f;

#define PP_            196
#define EPI            38416   // 196*196 elements per image
#define NGI            9604u   // 196*196/4 float4-groups per image
#define GROUPS_PER_ROW 49      // 196/4
#define WORDS_PER_IMG  7       // ceil(196/32)
#define MAIN_THREADS   256
#define NITER          38u     // ceil(9604/256)
#define DIAG_BLOCKS    784     // 1024*196 / 256 exactly

// ---------------- kernel 1: per-image streaming loss, WMMA-accumulated ------
// One block per image: b = blockIdx.x is scalar -> SGPR base addresses, no
// division by 9604, mask row staged in LDS.
__global__ __launch_bounds__(MAIN_THREADS) void cc_loss_main(
    const float* __restrict__ real, const float* __restrict__ fake,
    const float* __restrict__ fw, float* __restrict__ partials) {
  const unsigned tid = threadIdx.x;
  const unsigned b   = blockIdx.x;

  // ---- stage this image's 196-bit fake mask into 7 LDS words ----
  __shared__ unsigned mrow[WORDS_PER_IMG];
  __shared__ float    smem[8];
  if (tid < WORDS_PER_IMG) mrow[tid] = 0u;
  __syncthreads();
  if (tid < PP_) {
    if (fw[b * PP_ + tid] > 0.0f)
      atomicOr(&mrow[tid >> 5], 1u << (tid & 31u));
  }
  __syncthreads();

  const v4f* __restrict__ real4 = (const v4f*)(real + (size_t)b * EPI);
  const v4f* __restrict__ fake4 = (const v4f*)(fake + (size_t)b * EPI);

  v8f acc = {};                               // WMMA C/D accumulator
  v2f bones; bones.x = 1.0f; bones.y = 1.0f;  // B = ones -> D = row-sums of A

  for (unsigned it = 0; it < NITER; ++it) {
    unsigned g = it * MAIN_THREADS + tid;
    float sr = 0.0f, sf = 0.0f;
    if (g < NGI) {
      unsigned i  = g / GROUPS_PER_ROW;
      unsigned j0 = (g - i * GROUPS_PER_ROW) << 2;

      v4f r4 = __builtin_nontemporal_load(&real4[g]);
      v4f f4 = __builtin_nontemporal_load(&fake4[g]);

      // real: target is 0.8 everywhere (diagonal fixed up by cc_diag_fix)
      sr = fabsf(r4.x - 0.8f) + fabsf(r4.y - 0.8f) +
           fabsf(r4.z - 0.8f) + fabsf(r4.w - 0.8f);

      // fake: |f - cf|, cf = 0.8 - 0.7*(mask_i ^ mask_j)
      //     = |(f - 0.8) + 0.7*xorbit|
      unsigned mi = (mrow[i  >> 5] >> (i  & 31u)) & 1u;
      unsigned xw = ((mrow[j0 >> 5] >> (j0 & 31u)) ^ (0u - mi)) & 0xFu;
      float d0 = (f4.x - 0.8f) + 0.7f * (float)( xw        & 1u);
      float d1 = (f4.y - 0.8f) + 0.7f * (float)((xw >> 1u) & 1u);
      float d2 = (f4.z - 0.8f) + 0.7f * (float)((xw >> 2u) & 1u);
      float d3 = (f4.w - 0.8f) + 0.7f * (float)((xw >> 3u) & 1u);
      sf = fabsf(d0) + fabsf(d1) + fabsf(d2) + fabsf(d3);
    }
    // WMMA reduction with full EXEC; tail lanes feed exact zeros.
    v2f a; a.x = sr; a.y = sf;
    acc = __builtin_amdgcn_wmma_f32_16x16x4_f32(
        false, a, false, bones, (short)0, acc, false, false);
  }

  // Sum(D over wave) == 16 * Sum(all A values fed in).
  float s = acc[0] + acc[1] + acc[2] + acc[3] +
            acc[4] + acc[5] + acc[6] + acc[7];
#pragma unroll
  for (int off = 16; off > 0; off >>= 1) s += __shfl_xor(s, off, 32);

  int lane = tid & 31, wv = tid >> 5;
  if (lane == 0) smem[wv] = s;
  __syncthreads();
  if (wv == 0) {
    float t = (lane < 8) ? smem[lane] : 0.0f;
#pragma unroll
    for (int off = 4; off > 0; off >>= 1) t += __shfl_xor(t, off, 32);
    if (lane == 0) partials[blockIdx.x] = t;   // NOTE: 16x inflated
  }
}

// ---------------- kernel 2: diagonal correction for the real loss -----------
// Adds sum over diagonal of (|r-1.0| - |r-0.8|); 1024*196 = 784*256 threads.
__global__ __launch_bounds__(256) void cc_diag_fix(
    const float* __restrict__ real, float* __restrict__ pdiag) {
  int t = blockIdx.x * 256 + threadIdx.x;     // 0 .. 200703
  int b = t / PP_;
  int i = t - b * PP_;
  float x = real[(size_t)b * EPI + (size_t)i * (PP_ + 1)];
  float s = fabsf(x - 1.0f) - fabsf(x - 0.8f);
#pragma unroll
  for (int off = 16; off > 0; off >>= 1) s += __shfl_xor(s, off, 32);
  __shared__ float smem[8];
  int lane = threadIdx.x & 31, wv = threadIdx.x >> 5;
  if (lane == 0) smem[wv] = s;
  __syncthreads();
  if (wv == 0) {
    float u = (lane < 8) ? smem[lane] : 0.0f;
#pragma unroll
    for (int off = 4; off > 0; off >>= 1) u += __shfl_xor(u, off, 32);
    if (lane == 0) pdiag[blockIdx.x] = u;     // direct (not inflated)
  }
}

// ---------------- kernel 3: final deterministic reduce ----------------------
__global__ __launch_bounds__(256) void cc_final_reduce(
    const float* __restrict__ pmain, int nmain,
    const float* __restrict__ pdiag, int ndiag,
    float* __restrict__ out) {
  float sm = 0.0f, sd = 0.0f;
  for (int i = threadIdx.x; i < nmain; i += 256) sm += pmain[i];
  for (int i = threadIdx.x; i < ndiag; i += 256) sd += pdiag[i];
  float s = sm * (1.0f / 16.0f) + sd;   // undo WMMA 16x replication
#pragma unroll
  for (int off = 16; off > 0; off >>= 1) s += __shfl_xor(s, off, 32);
  __shared__ float smem[8];
  int lane = threadIdx.x & 31, wv = threadIdx.x >> 5;
  if (lane == 0) smem[wv] = s;
  __syncthreads();
  if (threadIdx.x == 0) {
    float t = 0.0f;
    for (int k = 0; k < 8; ++k) t += smem[k];
    // denom = B*(PP*PP - PP) = 1024*38220
    out[0] = t * (1.0f / 39137280.0f);
  }
}

extern "C" void kernel_launch(void* const* d_in, const int* in_sizes, int n_in,
                              void* d_out, int out_size, void* d_ws, size_t ws_size,
                              hipStream_t stream) {
  const float* real = (const float*)d_in[0];
  const float* fake = (const float*)d_in[1];
  const float* fw   = (const float*)d_in[2];
  float* out = (float*)d_out;

  const int B = in_sizes[2] / (14 * 14);            // 1024 images

  float* pmain = (float*)d_ws;                      // B floats
  float* pdiag = (float*)((char*)d_ws + 16384);     // DIAG_BLOCKS floats

  cc_loss_main<<<B, MAIN_THREADS, 0, stream>>>(real, fake, fw, pmain);
  cc_diag_fix<<<DIAG_BLOCKS, 256, 0, stream>>>(real, pdiag);
  cc_final_reduce<<<1, 256, 0, stream>>>(pmain, B, pdiag, DIAG_BLOCKS, out);
}